// SoftPolygonBatch_4595615007275
// MI455X (gfx1250) — compile-verified
//
#include <hip/hip_runtime.h>

// SoftPolygon on MI455X (gfx1250, wave32).
//
// The per-(pixel,edge) math decomposes into four affine forms in (px,py,1,|p|^2):
//   cross   = -esy*px + esx*py + (esy*vsx - esx*vsy)           (winding sign)
//   dot     =  esx*px + esy*py - (es.vs)                        (edge parameter)
//   dist_b  = -2vsx*px - 2vsy*py + |vs|^2 + |p|^2               (|p-vs|^2)
//   dist_c  = -2vex*px - 2vey*py + |ve|^2 + |p|^2               (|p-ve|^2)
// evaluated on the matrix pipe with V_WMMA_F32_16X16X4_F32
// (A = 16 edges x 4 coefs, B = 4 x 16 pixels; K=3 row carries |p|^2 so the
// distance accumulators come out exact). dist_a = dist_b - dot*param.
// Branchy select / min / winding stays on VALU and co-executes with WMMA
// (XDL WMMA is tracked as TRANS on CDNA5).
//
// Winding: cond_a = py > vs_y, cond_b = py < ve_y depend only on (row, edge),
// so a signed selector s(row,edge) in {+1,-1,0} is precomputed in LDS and the
// per-element update is  wn += (cross*s > 0) ? s : 0.
//
// Layout per CDNA5 ISA 7.12.2:
//   A(16x4 f32, 2 VGPR): lanes 0-15 hold K=0,1 ; lanes 16-31 hold K=2,3 ; M = lane&15
//   B(4x16 f32, 2 VGPR): mirrored (lane = column N)
//   D(16x16 f32, 8 VGPR): lane&15 = column N (pixel), VGPR r = row M=r (lo) / 8+r (hi)

typedef __attribute__((ext_vector_type(2))) float v2f;
typedef __attribute__((ext_vector_type(8))) float v8f;

#define POLY_V 64
#define IMG_W  64
#define IMG_H  64
#define NPIX   (IMG_W * IMG_H)          // 4096 pixels
#define NTILES (NPIX / 16)              // 256 pixel tiles of 16
#define SLICES 8                        // pixel-slices per polygon (grid.y)
#define ROWS_PER_BLOCK (IMG_H / SLICES) // 8 image rows per block
#define WAVES_PER_BLOCK 8
#define BLOCK_THREADS   256

__global__ __launch_bounds__(BLOCK_THREADS)
void SoftPolygonBatch_kernel(const float* __restrict__ verts,
                             float* __restrict__ out)
{
    // Coefficients for the A matrices: [type][k][edge]
    // type 0: cross, 1: dot, 2: dist_b, 3: dist_c   (k = px, py, const; K=3 coef is implicit)
    __shared__ float  sCoef[4][3][POLY_V];
    __shared__ float  sInv[POLY_V];                    // 1 / (|es|^2 + 1e-5)
    __shared__ float  sVsy[POLY_V];
    __shared__ float  sVey[POLY_V];
    __shared__ float2 sIS[ROWS_PER_BLOCK][POLY_V];     // {inv_sq_len, winding selector s}

    const int n   = blockIdx.x;
    const int tid = threadIdx.x;

    if (tid < POLY_V) {
        const float2 vs = ((const float2*)verts)[n * POLY_V + tid];
        const float2 ve = ((const float2*)verts)[n * POLY_V + ((tid + POLY_V - 1) & (POLY_V - 1))];
        const float esx = ve.x - vs.x;
        const float esy = ve.y - vs.y;
        sCoef[0][0][tid] = -esy;
        sCoef[0][1][tid] =  esx;
        sCoef[0][2][tid] =  esy * vs.x - esx * vs.y;
        sCoef[1][0][tid] =  esx;
        sCoef[1][1][tid] =  esy;
        sCoef[1][2][tid] = -(esx * vs.x + esy * vs.y);
        sCoef[2][0][tid] = -2.0f * vs.x;
        sCoef[2][1][tid] = -2.0f * vs.y;
        sCoef[2][2][tid] =  vs.x * vs.x + vs.y * vs.y;
        sCoef[3][0][tid] = -2.0f * ve.x;
        sCoef[3][1][tid] = -2.0f * ve.y;
        sCoef[3][2][tid] =  ve.x * ve.x + ve.y * ve.y;
        sInv[tid] = 1.0f / (esx * esx + esy * esy + 1e-5f);
        sVsy[tid] = vs.y;
        sVey[tid] = ve.y;
    }
    __syncthreads();

    // Winding selector per (local image row, edge).
    for (int idx = tid; idx < ROWS_PER_BLOCK * POLY_V; idx += BLOCK_THREADS) {
        const int   lr = idx >> 6;
        const int   e  = idx & (POLY_V - 1);
        const float py = (float)(blockIdx.y * ROWS_PER_BLOCK + lr);
        const bool  ca = py > sVsy[e];                 // pvs_y > 0
        const bool  cb = py < sVey[e];                 // py - ve_y < 0
        const float s  = (ca && cb) ? 1.0f : ((!ca && !cb) ? -1.0f : 0.0f);
        sIS[lr][e] = make_float2(sInv[e], s);
    }
    __syncthreads();

    const int  lane   = tid & 31;
    const int  wave   = tid >> 5;
    const int  col    = lane & 15;
    const bool hiHalf = lane >= 16;

    // Resident A operands: [type][edge-tile], 32 VGPRs, constant per block.
    v2f A[4][4];
#pragma unroll
    for (int t = 0; t < 4; ++t) {
        const int e = t * 16 + col;
#pragma unroll
        for (int ty = 0; ty < 4; ++ty) {
            const float k0 = sCoef[ty][0][e];
            const float k1 = sCoef[ty][1][e];
            const float k2 = sCoef[ty][2][e];
            v2f a;
            a.x = hiHalf ? k2 : k0;                            // K=2 (const) : K=0 (px)
            a.y = hiHalf ? ((ty >= 2) ? 1.0f : 0.0f) : k1;     // K=3 (|p|^2 gate) : K=1 (py)
            A[ty][t] = a;
        }
    }

    float* outp = out + (size_t)n * NPIX;

    const int tile0 = blockIdx.y * (NTILES / SLICES);
    for (int i = 0; i < (NTILES / SLICES) / WAVES_PER_BLOCK; ++i) {
        const int tloc = wave + WAVES_PER_BLOCK * i;   // tile within block slice
        const int tt   = tile0 + tloc;
        const int base = tt * 16;                      // 16 pixels share one image row
        const int lr   = tloc >> 2;                    // local row (4 tiles per row)
        const float py  = (float)(base >> 6);
        const float px  = (float)((base & 63) + col);
        const float pp2 = px * px + py * py;

        v2f B;
        B.x = hiHalf ? 1.0f : px;                      // K=2 : K=0
        B.y = hiHalf ? pp2  : py;                      // K=3 : K=1

        float dmin = 3.4e38f;
        float wnf  = 0.0f;
        const v8f zero = {};

#pragma unroll
        for (int t = 0; t < 4; ++t) {
            v8f accC = __builtin_amdgcn_wmma_f32_16x16x4_f32(false, A[0][t], false, B, (short)0, zero, false, false);
            v8f accD = __builtin_amdgcn_wmma_f32_16x16x4_f32(false, A[1][t], false, B, (short)0, zero, false, false);
            v8f accB = __builtin_amdgcn_wmma_f32_16x16x4_f32(false, A[2][t], false, B, (short)0, zero, false, false);
            v8f accP = __builtin_amdgcn_wmma_f32_16x16x4_f32(false, A[3][t], false, B, (short)0, zero, false, false);

            const int ebase = t * 16 + (hiHalf ? 8 : 0);   // edge row of acc element r
#pragma unroll
            for (int r = 0; r < 8; ++r) {
                const float2 is = sIS[lr][ebase + r];      // broadcast ds_load_b64
                const float cr = accC[r];
                const float dt = accD[r];
                const float db = accB[r];                  // dist_b (|p|^2 folded in)
                const float dp = accP[r];                  // dist_c
                const float param = dt * is.x;
                const float da = fmaf(-dt, param, db);     // dist_a
                const float dist = (param < 0.0f) ? db : ((param > 1.0f) ? dp : da);
                dmin = fminf(dmin, dist);
                wnf += (cr * is.y > 0.0f) ? is.y : 0.0f;   // winding contribution
            }
        }

        // Combine the two wave halves (the other 32 edges live in partner lane).
        dmin = fminf(dmin, __shfl_xor(dmin, 16, 32));
        wnf += __shfl_xor(wnf, 16, 32);

        if (!hiHalf) {
            const float inside = (wnf != 0.0f) ? 1.0f : -1.0f;
            const float x = dmin * inside * 10.0f;         // / INV_SMOOTHNESS
            outp[base + col] = 1.0f / (1.0f + __expf(-x));
        }
    }
}

extern "C" void kernel_launch(void* const* d_in, const int* in_sizes, int n_in,
                              void* d_out, int out_size, void* d_ws, size_t ws_size,
                              hipStream_t stream) {
    const float* verts = (const float*)d_in[0];
    const int nPoly = in_sizes[0] / (POLY_V * 2);   // 128 for the reference shapes
    dim3 grid(nPoly, SLICES);
    SoftPolygonBatch_kernel<<<grid, BLOCK_THREADS, 0, stream>>>(verts, (float*)d_out);
}